// CustomOrientationLoss_36447092474035
// MI455X (gfx1250) — compile-verified
//
#include <hip/hip_runtime.h>
#include <hip/hip_bf16.h>
#include <math.h>

typedef __attribute__((ext_vector_type(2))) float v2f;
typedef __attribute__((ext_vector_type(8))) float v8f;

// ---------------------------------------------------------------------------
// Workspace layout: Z[N*3] accumulator, then accum[4] = {sum_loss, sum_ang, cnt, pad}
// ---------------------------------------------------------------------------

__global__ void col_zero_kernel(float* __restrict__ p, int n) {
    int i = blockIdx.x * blockDim.x + threadIdx.x;
    if (i < n) p[i] = 0.0f;
}

__device__ __forceinline__ float atomicAddAgent(float* p, float v) {
    return __hip_atomic_fetch_add(p, v, __ATOMIC_RELAXED, __HIP_MEMORY_SCOPE_AGENT);
}

// One thread per undirected edge: contribution (x[v]-x[u])*(s[v]-s[u]) is the
// same for both endpoints, so compute once, scatter to both.
__global__ void col_edge_scatter_kernel(const int* __restrict__ eidx,
                                        const float* __restrict__ x,
                                        const float* __restrict__ s,
                                        float* __restrict__ Z,
                                        int E) {
    int e = blockIdx.x * blockDim.x + threadIdx.x;
    if (e >= E) return;
    int u = eidx[e];
    int v = eidx[e + E];
    float dsv = s[v] - s[u];
    float d0 = (x[3 * v + 0] - x[3 * u + 0]) * dsv;
    float d1 = (x[3 * v + 1] - x[3 * u + 1]) * dsv;
    float d2 = (x[3 * v + 2] - x[3 * u + 2]) * dsv;
    atomicAddAgent(&Z[3 * u + 0], d0);
    atomicAddAgent(&Z[3 * u + 1], d1);
    atomicAddAgent(&Z[3 * u + 2], d2);
    atomicAddAgent(&Z[3 * v + 0], d0);
    atomicAddAgent(&Z[3 * v + 1], d1);
    atomicAddAgent(&Z[3 * v + 2], d2);
}

// One wave handles 16 nodes: y.Z dot products for the 16 nodes come from the
// diagonal of a single V_WMMA_F32_16X16X4_F32 (A rows = y, B cols = Z).
__global__ void col_node_loss_kernel(const float* __restrict__ y,
                                     const float* __restrict__ Z,
                                     const int* __restrict__ mask,
                                     float* __restrict__ accum,
                                     int N) {
    const int lane = threadIdx.x & 31;
    const int waveInBlock = threadIdx.x >> 5;
    const int wavesPerBlock = blockDim.x >> 5;
    const int tile = blockIdx.x * wavesPerBlock + waveInBlock;
    const int tiles = (N + 15) >> 4;
    if (tile >= tiles) return;  // wave-uniform: EXEC stays all-ones for live waves

    const int m = lane & 15;   // node within tile (cols; rows for A)
    const int hi = lane >> 4;  // half-wave select (K split / row-block select)
    const int n = tile * 16 + m;
    const bool valid = n < N;

    float y0 = 0.f, y1 = 0.f, y2 = 0.f, z0 = 0.f, z1 = 0.f, z2 = 0.f;
    if (valid) {
        y0 = y[3 * n + 0]; y1 = y[3 * n + 1]; y2 = y[3 * n + 2];
        z0 = Z[3 * n + 0]; z1 = Z[3 * n + 1]; z2 = Z[3 * n + 2];
    }

    // A (16x4 f32): lanes 0-15 hold K=0(V0),K=1(V1); lanes 16-31 hold K=2(V0),K=3(V1).
    v2f a;
    a.x = hi ? y2 : y0;
    a.y = hi ? 0.f : y1;  // K=3 pad
    // B (4x16 f32): V0 = row0 (lanes 0-15) / row2 (lanes 16-31); V1 = row1 / row3.
    v2f b;
    b.x = hi ? z2 : z0;
    b.y = hi ? 0.f : z1;  // K=3 pad

    v8f c = {};
    c = __builtin_amdgcn_wmma_f32_16x16x4_f32(
        /*neg_a=*/false, a, /*neg_b=*/false, b,
        /*c_mod=*/(short)0, c, /*reuse_a=*/false, /*reuse_b=*/false);

    // D diagonal: node m<8 -> lane m, VGPR m; node m>=8 -> lane m+16, VGPR m-8.
    const bool diagHolder = (hi == 0) ? (m < 8) : (m >= 8);

    float lsum = 0.f, asum = 0.f, csum = 0.f;
    if (diagHolder && valid) {
        int vi = hi ? (m - 8) : m;
        float dot = c[vi];  // y[n] . Z[n]
        float ny = sqrtf(y0 * y0 + y1 * y1 + y2 * y2);
        float nz = sqrtf(z0 * z0 + z1 * z1 + z2 * z2);
        float cosv = dot / (ny * nz);
        float mk = (mask[n] != 0) ? 1.0f : 0.0f;
        float cl = fminf(fmaxf(cosv, -1.0f), 1.0f);
        float ang = acosf(cl) * 57.29577951308232f;  // degrees
        lsum = mk * (1.0f - fabsf(cosv));
        asum = mk * ang;
        csum = mk;
    }

    // Full-wave tree reduction (inactive lanes contribute 0).
    for (int off = 16; off > 0; off >>= 1) {
        lsum += __shfl_down(lsum, off, 32);
        asum += __shfl_down(asum, off, 32);
        csum += __shfl_down(csum, off, 32);
    }
    if (lane == 0) {
        atomicAddAgent(&accum[0], lsum);
        atomicAddAgent(&accum[1], asum);
        atomicAddAgent(&accum[2], csum);
    }
}

__global__ void col_finalize_kernel(const float* __restrict__ accum,
                                    float* __restrict__ out) {
    if (threadIdx.x == 0 && blockIdx.x == 0) {
        float cnt = fmaxf(accum[2], 1.0f);
        out[0] = accum[0] / cnt;
        out[1] = accum[1] / cnt;
    }
}

extern "C" void kernel_launch(void* const* d_in, const int* in_sizes, int n_in,
                              void* d_out, int out_size, void* d_ws, size_t ws_size,
                              hipStream_t stream) {
    (void)n_in; (void)out_size; (void)ws_size;
    const float* x    = (const float*)d_in[0];           // [N,3]
    const float* y    = (const float*)d_in[1];           // [N,3]
    const float* s    = (const float*)d_in[2];           // [N]
    const int*   eidx = (const int*)d_in[3];             // [2,E] flattened
    const int*   mask = (const int*)d_in[4];             // [N]

    const int N = in_sizes[0] / 3;
    const int E = in_sizes[3] / 2;

    float* Z = (float*)d_ws;                 // N*3 floats
    float* accum = Z + (size_t)N * 3;        // 4 floats

    const int zn = N * 3 + 4;
    col_zero_kernel<<<(zn + 255) / 256, 256, 0, stream>>>(Z, zn);

    col_edge_scatter_kernel<<<(E + 255) / 256, 256, 0, stream>>>(eidx, x, s, Z, E);

    const int tiles = (N + 15) / 16;
    const int wavesPerBlock = 8;  // 256 threads = 8 wave32
    const int blocks = (tiles + wavesPerBlock - 1) / wavesPerBlock;
    col_node_loss_kernel<<<blocks, 256, 0, stream>>>(y, Z, mask, accum, N);

    col_finalize_kernel<<<1, 64, 0, stream>>>(accum, (float*)d_out);
}